// ModelLayer_13116830122424
// MI455X (gfx1250) — compile-verified
//
#include <hip/hip_runtime.h>
#include <cstdint>

// ---------------------------------------------------------------------------
// Problem constants (from reference): N=4096, T=256, SIZE=8, K=16
// ---------------------------------------------------------------------------
#define NN        4096
#define TT        256
#define KCUT      16
#define TOUT      (TT - KCUT)           // 240
#define ROWSTRIDE 4100                  // padded LDS row stride (dwords); 4100%64=4 -> conflict-free
#define GRID      256                   // one 16-row tile of M per workgroup
#define LDS_FLOATS (16 * ROWSTRIDE + 8 * 16)
#define LDS_BYTES  (LDS_FLOATS * 4)     // 262,912 B <= 320 KB WGP LDS

typedef __attribute__((ext_vector_type(2))) float v2f;
typedef __attribute__((ext_vector_type(8))) float v8f;

// D = A(16x4) * B(4x16) + C(16x16), all f32 (CDNA5 V_WMMA_F32_16X16X4_F32)
static __device__ __forceinline__ v8f wmma4(float ax, float ay,
                                            float bx, float by, v8f c) {
  v2f a; a.x = ax; a.y = ay;
  v2f b; b.x = bx; b.y = by;
  return __builtin_amdgcn_wmma_f32_16x16x4_f32(false, a, false, b,
                                               (short)0, c, false, false);
}

// ---------------------------------------------------------------------------
// Phase 0a: tempt[t][n] = sum_s x[n][t][s] * weight[s]   (stored [T][N])
// ---------------------------------------------------------------------------
__global__ __launch_bounds__(256) void tempt_kernel(
    const float* __restrict__ x, const float* __restrict__ w,
    float* __restrict__ tempt) {
  int idx = blockIdx.x * 256 + threadIdx.x;   // idx = n*T + t  (matches x layout)
  int n = idx >> 8;
  int t = idx & (TT - 1);
  const float4* xp = (const float4*)(x + (size_t)idx * 8);
  float4 x0 = xp[0];
  float4 x1 = xp[1];
  float acc = x0.x * w[0] + x0.y * w[1] + x0.z * w[2] + x0.w * w[3] +
              x1.x * w[4] + x1.y * w[5] + x1.z * w[6] + x1.w * w[7];
  tempt[(size_t)t * NN + n] = acc;
}

// ---------------------------------------------------------------------------
// Phase 0b: hs_init[i] = exp(xi) * hidden[i]  (pre-scaled initial state)
// ---------------------------------------------------------------------------
__global__ __launch_bounds__(256) void hscale_kernel(
    const float* __restrict__ hidden, const float* __restrict__ xi,
    float* __restrict__ hs_init) {
  int i = blockIdx.x * 256 + threadIdx.x;
  hs_init[i] = __expf(xi[0]) * hidden[i];
}

// ---------------------------------------------------------------------------
// Phase 1: persistent scan kernel.
// Block b owns output rows [b*16, b*16+16). Its 16x4096 f32 tile of M lives in
// LDS for all 256 steps (loaded once via async global->LDS copies). The state
// buffers hold the PRE-SCALED vector hs_t = exp(xi)*h_t so the inner loop is
// pure load+WMMA (no VALU on the operand path). Each of the 8 waves reduces a
// 512-wide j-slice with f32 WMMA on 4 independent accumulator chains;
// partials are combined in LDS, tempt added, outputs written, and all 256
// blocks sync with an agent-scope sense-reversal barrier between steps.
// ---------------------------------------------------------------------------
__global__ __launch_bounds__(256) void step_kernel(
    const float* __restrict__ M, const float* __restrict__ hs_init,
    const float* __restrict__ xi, const float* __restrict__ tempt,
    float* __restrict__ hs0, float* __restrict__ hs1,
    float* __restrict__ out, int* bar_cnt, int* bar_gen) {
  extern __shared__ float lds[];
  float* ldsM = lds;                        // 16 x ROWSTRIDE (padded rows)
  float* red  = lds + 16 * ROWSTRIDE;       // 8 waves x 16 partials

  const int tid  = threadIdx.x;
  const int lane = tid & 31;
  const int wav  = tid >> 5;
  const int i0   = blockIdx.x * 16;

  // ---- One-time: async-copy M rows i0..i0+15 into LDS (ASYNCcnt path) ----
  for (int it = 0; it < 64; ++it) {
    int c    = it * 256 + tid;              // 16384 x 16B chunks
    int row  = c >> 10;
    int col4 = c & 1023;
    unsigned ldsoff = (unsigned)(uint64_t)(ldsM + row * ROWSTRIDE + col4 * 4);
    const float* gp = M + ((size_t)(i0 + row) * NN + col4 * 4);
    asm volatile("global_load_async_to_lds_b128 %0, %1, off"
                 :: "v"(ldsoff), "v"(gp) : "memory");
  }
  asm volatile("s_wait_asynccnt 0" ::: "memory");
  __syncthreads();

  const float sc   = __expf(xi[0]);         // only used in the 16-lane epilogue
  const int   nsel = lane & 15;             // WMMA N (output row) / A M-index
  const int   koff = (lane >> 4) << 1;      // K base for this half-wave: 0 or 2
  const float* brow = ldsM + nsel * ROWSTRIDE + koff;
  const int jbase = wav * 512;

  for (int t = 0; t < TT; ++t) {
    const float* hsrc = (t == 0) ? hs_init : ((t & 1) ? hs0 : hs1);
    float*       hdst = (t & 1) ? hs1 : hs0;
    const float* hp   = hsrc + koff;

    v8f a0 = {}, a1 = {}, a2 = {}, a3 = {};  // 4 chains hide WMMA latency
#pragma unroll 4
    for (int j = jbase; j < jbase + 512; j += 16) {
      float2 h0 = *(const float2*)(hp + j);
      float2 h1 = *(const float2*)(hp + j + 4);
      float2 h2 = *(const float2*)(hp + j + 8);
      float2 h3 = *(const float2*)(hp + j + 12);
      float2 b0 = *(const float2*)(brow + j);
      float2 b1 = *(const float2*)(brow + j + 4);
      float2 b2 = *(const float2*)(brow + j + 8);
      float2 b3 = *(const float2*)(brow + j + 12);
      a0 = wmma4(h0.x, h0.y, b0.x, b0.y, a0);
      a1 = wmma4(h1.x, h1.y, b1.x, b1.y, a1);
      a2 = wmma4(h2.x, h2.y, b2.x, b2.y, a2);
      a3 = wmma4(h3.x, h3.y, b3.x, b3.y, a3);
    }
    // C VGPR0, lanes 0-15 hold y_partial[i0 + n] (broadcast over M rows)
    float part = a0[0] + a1[0] + a2[0] + a3[0];
    if (lane < 16) red[wav * 16 + nsel] = part;
    __syncthreads();

    if (tid < 16) {
      float y = 0.0f;
#pragma unroll
      for (int wv = 0; wv < 8; ++wv) y += red[wv * 16 + tid];
      y += tempt[(size_t)t * NN + i0 + tid];        // y = h_t[i0+tid]
      hdst[i0 + tid] = sc * y;                      // store pre-scaled state
      if (t >= KCUT)   out[(size_t)(i0 + tid) * TOUT + (t - KCUT)] = y;
      if (t == TT - 1) out[(size_t)NN * TOUT + i0 + tid] = y;
    }
    __threadfence();

    // ---- grid-wide sense-reversal barrier (agent scope) ----
    __syncthreads();
    if (tid == 0) {
      int g = __hip_atomic_load(bar_gen, __ATOMIC_ACQUIRE,
                                __HIP_MEMORY_SCOPE_AGENT);
      if (__hip_atomic_fetch_add(bar_cnt, 1, __ATOMIC_ACQ_REL,
                                 __HIP_MEMORY_SCOPE_AGENT) == (int)gridDim.x - 1) {
        __hip_atomic_store(bar_cnt, 0, __ATOMIC_RELAXED,
                           __HIP_MEMORY_SCOPE_AGENT);
        __hip_atomic_fetch_add(bar_gen, 1, __ATOMIC_ACQ_REL,
                               __HIP_MEMORY_SCOPE_AGENT);
      } else {
        while (__hip_atomic_load(bar_gen, __ATOMIC_ACQUIRE,
                                 __HIP_MEMORY_SCOPE_AGENT) == g)
          __builtin_amdgcn_s_sleep(2);
      }
    }
    __syncthreads();
  }
}

// ---------------------------------------------------------------------------
// Host launcher
// ---------------------------------------------------------------------------
extern "C" void kernel_launch(void* const* d_in, const int* in_sizes, int n_in,
                              void* d_out, int out_size, void* d_ws,
                              size_t ws_size, hipStream_t stream) {
  (void)in_sizes; (void)n_in; (void)out_size; (void)ws_size;
  const float* x      = (const float*)d_in[0];  // [N,T,8]
  const float* hidden = (const float*)d_in[1];  // [N]
  const float* M      = (const float*)d_in[2];  // [N,N]
  const float* weight = (const float*)d_in[3];  // [8]
  const float* xi     = (const float*)d_in[4];  // [1]
  // d_in[5] is k == 16, baked in as KCUT.
  float* out = (float*)d_out;

  float* tempt  = (float*)d_ws;                 // T*N floats (4 MB)
  float* hs0    = tempt + (size_t)NN * TT;      // N floats (pre-scaled state)
  float* hs1    = hs0 + NN;                     // N floats
  float* hsinit = hs1 + NN;                     // N floats
  int*   bar    = (int*)(hsinit + NN);          // {cnt, gen}

  hipMemsetAsync(bar, 0, 2 * sizeof(int), stream);
  tempt_kernel<<<(NN * TT) / 256, 256, 0, stream>>>(x, weight, tempt);
  hscale_kernel<<<NN / 256, 256, 0, stream>>>(hidden, xi, hsinit);

  hipFuncSetAttribute(reinterpret_cast<const void*>(step_kernel),
                      hipFuncAttributeMaxDynamicSharedMemorySize, LDS_BYTES);
  step_kernel<<<GRID, 256, LDS_BYTES, stream>>>(M, hsinit, xi, tempt, hs0, hs1,
                                                out, bar, bar + 1);
}